// WritingCell_83227876262060
// MI455X (gfx1250) — compile-verified
//
#include <hip/hip_runtime.h>

typedef __attribute__((ext_vector_type(2))) float v2f;
typedef __attribute__((ext_vector_type(8))) float v8f;

#define B_  16
#define T_  64
#define U_  256

// One wave32 owns one 16x16 tile of mem for one batch.
// Per step: two V_WMMA_F32_16X16X4_F32 outer products (k (x) v, k^2 (x) 1),
// then a 4-op/element VALU recurrence, then 8 coalesced b32 stores.
__global__ __launch_bounds__(256) void writing_cell_kernel(
    const float* __restrict__ x,     // (B, T, 2U)
    const float* __restrict__ gpos,  // (1,)
    const float* __restrict__ gneg,  // (1,)
    float* __restrict__ out)         // (B, T, U, U)
{
    const float gp = gpos[0];
    const float gn = gneg[0];

    const int lane = threadIdx.x & 31;
    const int wave = blockIdx.x * (blockDim.x >> 5) + (threadIdx.x >> 5);

    // 4096 waves: 16 batches x 256 tiles (16x16 tiles of the 256x256 mem)
    const int b    = wave >> 8;
    const int tile = wave & 255;
    const int it   = tile >> 4;   // row-tile index
    const int jt   = tile & 15;   // col-tile index

    const bool lo  = lane < 16;
    const int  l15 = lane & 15;
    const int  hi  = lane >> 4;   // 0: rows 0..7, 1: rows 8..15 (C/D layout)
    const int  n   = l15;         // column within tile held by this lane

    v8f mem = {};                 // 16x16 f32 accumulator tile (8 VGPRs)

    const float* xb = x + (size_t)b * T_ * (2 * U_);
    float* ob = out + (size_t)b * T_ * U_ * U_
                    + (size_t)(it * 16) * U_ + (size_t)(jt * 16);

    // B operand for the k^2 outer product: row K=0 is all-ones (loop invariant)
    v2f Bones = { lo ? 1.0f : 0.0f, 0.0f };

    for (int t = 0; t < T_; ++t) {
        const float* xt = xb + t * (2 * U_);

        // lanes 0..15 carry the live values; upper half-wave zeroed so only
        // the K=0 slice of the 16x16x4 contraction is populated.
        float kv = xt[it * 16 + l15];        // k[i] for this tile, 64B coalesced
        float vv = xt[U_ + jt * 16 + l15];   // v[j] for this tile, 64B coalesced
        float ka = lo ? kv : 0.0f;
        float va = lo ? vv : 0.0f;

        v2f A1 = { ka, 0.0f };               // A: column K=0 = k
        v2f B1 = { va, 0.0f };               // B: row    K=0 = v
        v2f A2 = { ka * ka, 0.0f };          // A: column K=0 = k*k

        v8f P = {};                          // P[m][n] = k[m]*v[n]
        v8f Q = {};                          // Q[m][n] = k[m]^2
        P = __builtin_amdgcn_wmma_f32_16x16x4_f32(
                false, A1, false, B1, (short)0, P, false, false);
        Q = __builtin_amdgcn_wmma_f32_16x16x4_f32(
                false, A2, false, Bones, (short)0, Q, false, false);

        float* orow = ob + (size_t)t * U_ * U_;
        #pragma unroll
        for (int r = 0; r < 8; ++r) {
            float p = gp * P[r];                       // gamma_pos * k[m]*v[n]
            float f = fmaf(-gn, Q[r], 1.0f - p);       // 1 - p - q
            float m = fmaf(mem[r], f, p);              // mem*(1-p-q) + p
            mem[r] = m;
            orow[(size_t)(r + 8 * hi) * U_ + n] = m;   // out[b,t,m,n]
        }
    }
}

extern "C" void kernel_launch(void* const* d_in, const int* in_sizes, int n_in,
                              void* d_out, int out_size, void* d_ws, size_t ws_size,
                              hipStream_t stream) {
    const float* x  = (const float*)d_in[0];
    const float* gp = (const float*)d_in[1];
    const float* gn = (const float*)d_in[2];
    float* out = (float*)d_out;

    // 4096 waves = 512 blocks x 8 waves (256 threads, multiple of wave32)
    dim3 grid(512), block(256);
    hipLaunchKernelGGL(writing_cell_kernel, grid, block, 0, stream,
                       x, gp, gn, out);
}